// LPCSynth_75608604278811
// MI455X (gfx1250) — compile-verified
//
#include <hip/hip_runtime.h>

// LPC synthesis filter + Hann overlap-add for MI455X (gfx1250, wave32).
// Blocked IIR: per 16-sample block, y = L * (g*x - M*s) done with chained
// v_wmma_f32_16x16x4_f32. One frame per wave, 8 waves (frames) per workgroup.

#define HOP      256
#define WINLEN   1024
#define PADN     384                    // (WINLEN-HOP)/2
#define ORDER    25
#define NFRAMES  4096
#define NX       (NFRAMES * HOP)        // 1048576 input samples
#define TOTALP   ((NFRAMES - 1) * HOP + WINLEN)  // 1049344 padded positions
#define NOUT     (TOTALP - 2 * PADN)    // 1048576 output samples
#define BLKLEN   16
#define NBLK     (WINLEN / BLKLEN)      // 64
#define WAVES_WG 8
// per-wave LDS floats: 32 zero-pad + 1024 y-history + 16 h + 16 r
#define PW_LDS   (32 + WINLEN + 16 + 16)

typedef float v2f __attribute__((ext_vector_type(2)));
typedef float v8f __attribute__((ext_vector_type(8)));

// Probe the builtin only in the DEVICE pass; the host clang never has it.
#if defined(__HIP_DEVICE_COMPILE__)
#if !__has_builtin(__builtin_amdgcn_wmma_f32_16x16x4_f32)
#error "gfx1250 wmma f32 16x16x4 builtin not available on device"
#endif
#endif

__device__ __forceinline__ float hannw(int t) {
  // periodic Hann: 0.5*(1 - cos(2*pi*t/1024)); 2*pi/1024 = 0.006135923151542565
  return 0.5f * (1.0f - __cosf(6.135923151542565e-3f * (float)t));
}

__global__ void lpc_zero(float* __restrict__ out) {
  int i = blockIdx.x * blockDim.x + threadIdx.x;
  if (i < NOUT) out[i] = 0.0f;
}

__global__ __launch_bounds__(256) void lpc_main(const float* __restrict__ ex,
                                                const float* __restrict__ lpc,
                                                float* __restrict__ out) {
  __shared__ float smem[PW_LDS * WAVES_WG];
  const int lane = threadIdx.x & 31;
  const int wv   = threadIdx.x >> 5;
  const int f    = blockIdx.x * WAVES_WG + wv;   // frame handled by this wave
  const int n    = lane & 15;                    // column index (N dim)
  const int bh   = lane >> 4;                    // lane half -> K sub-chunk

  float* ybuf = smem + wv * PW_LDS + 32;         // ybuf[t], t in [-32, WINLEN)
  float* hbuf = smem + wv * PW_LDS + 32 + WINLEN;
  float* rbuf = hbuf + 16;

  // zero the 32-float history pad (covers y[T-25..] for first blocks)
  ybuf[lane - 32] = 0.0f;

  const float* lp = lpc + f * (ORDER + 1);       // [gain, a1..a25]
  const float gain = lp[0];

  // ---- impulse response h[0..15] of the in-block recursion (taps <= 15) ----
  float a[16];
#pragma unroll
  for (int k = 1; k < 16; ++k) a[k] = lp[k];
  float h[16];
  h[0] = 1.0f;
#pragma unroll
  for (int nn = 1; nn < 16; ++nn) {
    float s = 0.0f;
#pragma unroll
    for (int k = 1; k <= nn; ++k) s += a[k] * h[nn - k];
    h[nn] = -s;
  }
  if (lane == 0) {
#pragma unroll
    for (int i = 0; i < 16; ++i) hbuf[i] = h[i];
  }
  // same-wave LDS ops are in-order; no barrier needed (wave-private region)

  // ---- per-frame B operands ----
  // WMMA#1 (state -> r): B[k][n] = -a[n+25-k] for n <= k <= 24, else 0
  //   (z_s[k] = y[T-25+k] so reversed-index coefficients)
  v2f BM[7];
#pragma unroll
  for (int c = 0; c < 7; ++c) {
    int k0 = 4 * c + 2 * bh;
    float v0 = 0.0f, v1 = 0.0f;
    if (k0 <= 24 && k0 >= n)         v0 = -lp[n + 25 - k0];
    if (k0 + 1 <= 24 && k0 + 1 >= n) v1 = -lp[n + 24 - k0];
    BM[c] = (v2f){v0, v1};
  }
  // WMMA#2 (r -> y): B[k][n] = L^T[k][n] = h[n-k] for n >= k, else 0
  v2f BL[4];
#pragma unroll
  for (int c = 0; c < 4; ++c) {
    int k0 = 4 * c + 2 * bh;
    float v0 = (n >= k0)     ? hbuf[n - k0]     : 0.0f;
    float v1 = (n >= k0 + 1) ? hbuf[n - k0 - 1] : 0.0f;
    BL[c] = (v2f){v0, v1};
  }

  // ---- 64 sequential blocks of 16 samples ----
  for (int blk = 0; blk < NBLK; ++blk) {
    const int T = blk * BLKLEN;

    // C matrix carries g*x broadcast down the rows: C[m][n] = g*x[T+n]
    int pex = f * HOP + T + n - PADN;
    float gx = (pex >= 0 && pex < NX) ? gain * ex[pex] : 0.0f;
    v8f acc = {gx, gx, gx, gx, gx, gx, gx, gx};

    // A rows broadcast z_s[k] = y[T-25+k] (k=0..24, rest zero)
#pragma unroll
    for (int c = 0; c < 7; ++c) {
      int k0 = 4 * c + 2 * bh;
      float z0 = (k0 <= 24)     ? ybuf[T - 25 + k0] : 0.0f;
      float z1 = (k0 + 1 <= 24) ? ybuf[T - 24 + k0] : 0.0f;
      v2f av = (v2f){z0, z1};
      acc = __builtin_amdgcn_wmma_f32_16x16x4_f32(
          false, av, false, BM[c], (short)0, acc, false, false);
    }
    // every lane's acc[0] == r[n]; park r in LDS for the next A operand
    if (lane < 16) rbuf[n] = acc[0];

    v8f acc2 = {0.f, 0.f, 0.f, 0.f, 0.f, 0.f, 0.f, 0.f};
#pragma unroll
    for (int c = 0; c < 4; ++c) {
      int k0 = 4 * c + 2 * bh;
      v2f av = (v2f){rbuf[k0], rbuf[k0 + 1]};
      acc2 = __builtin_amdgcn_wmma_f32_16x16x4_f32(
          false, av, false, BL[c], (short)0, acc2, false, false);
    }
    float yv = acc2[0];                          // y[T+n] on every lane

    if (lane < 16) {
      ybuf[T + n] = yv;                          // extend history
      int pp = f * HOP + T + n;                  // padded output position
      int oi = pp - PADN;
      if (oi >= 0 && oi < NOUT)
        atomicAdd(out + oi, yv * hannw(T + n));  // windowed overlap-add
    }
  }
}

__global__ void lpc_norm(float* __restrict__ out) {
  int i = blockIdx.x * blockDim.x + threadIdx.x;
  if (i >= NOUT) return;
  int p = i + PADN;
  int q = p & (HOP - 1);
  int fbase = p >> 8;                            // HOP == 256
  float norm = 0.0f;
#pragma unroll
  for (int r = 0; r < 4; ++r) {
    int fr = fbase - r;
    if (fr >= 0 && fr < NFRAMES) norm += hannw(q + HOP * r);
  }
  out[i] = out[i] / norm;
}

extern "C" void kernel_launch(void* const* d_in, const int* in_sizes, int n_in,
                              void* d_out, int out_size, void* d_ws, size_t ws_size,
                              hipStream_t stream) {
  (void)in_sizes; (void)n_in; (void)d_ws; (void)ws_size; (void)out_size;
  const float* ex  = (const float*)d_in[0];
  const float* lpc = (const float*)d_in[1];
  float* out = (float*)d_out;

  lpc_zero<<<(NOUT + 255) / 256, 256, 0, stream>>>(out);
  lpc_main<<<NFRAMES / WAVES_WG, 256, 0, stream>>>(ex, lpc, out);
  lpc_norm<<<(NOUT + 255) / 256, 256, 0, stream>>>(out);
}